// cheb_conv_withSAt_6640019439998
// MI455X (gfx1250) — compile-verified
//
#include <hip/hip_runtime.h>

typedef __attribute__((ext_vector_type(2))) float v2f;
typedef __attribute__((ext_vector_type(8))) float v8f;

#define BDIM 384
#define NWAVE 12

static constexpr int Bb = 16, Nn = 1024, Cin = 16, Tt = 36, Kk = 3, Cout = 64;
static constexpr int CT = Cin * Tt;            // 576
static constexpr int MB = 32;                  // m-chunk
static constexpr int XPITCH = 580;             // padded X row (keeps 16B align, kills conflicts)
static constexpr int TAT_PITCH = 34;           // padded t_at row (even -> b64 reads, conflict-free)
static constexpr int TH_PITCH = 50;            // padded theta^T row
static constexpr int RHS_TP = 40;              // padded t dimension of rhs

static constexpr int OFF_TAT = 0;                          // 3*16*34*4   = 6528 B
static constexpr int OFF_TH  = 6528;                       // 64*50*4     = 12800 B
static constexpr int OFF_BIG = 6528 + 12800;               // 19328 B
static constexpr int BIG_FLOATS = 16 * 48 * RHS_TP + 64;   // rhs union (>= X chunk), + slack
static constexpr int LDS_BYTES = OFF_BIG + BIG_FLOATS * 4; // 142464 B

__global__ __launch_bounds__(BDIM)
void cheb_conv_sat_kernel(const float* __restrict__ x,
                          const float* __restrict__ sat,
                          const float* __restrict__ cheb,
                          const float* __restrict__ theta,
                          float* __restrict__ out)
{
    extern __shared__ char smem[];
    float* tat = (float*)(smem + OFF_TAT);   // [3][16][TAT_PITCH]  = (cheb*SA)^T chunk
    float* thT = (float*)(smem + OFF_TH);    // [64][TH_PITCH]      = theta^T
    float* big = (float*)(smem + OFF_BIG);   // union: X[32][XPITCH] | rhs[16][48][RHS_TP]

    const int tid  = threadIdx.x;
    const int wave = tid >> 5;
    const int lane = tid & 31;
    const int half = lane >> 4;   // 0: lanes 0-15, 1: lanes 16-31
    const int nl   = lane & 15;

    const int b  = blockIdx.x >> 6;          // / 64
    const int n0 = (blockIdx.x & 63) << 4;   // * 16

    // theta[k,c,o] -> thT[o][kc]
    for (int e = tid; e < Kk * Cin * Cout; e += BDIM) {
        int kc = e >> 6;
        int o  = e & 63;
        thT[o * TH_PITCH + kc] = theta[e];
    }

    v8f acc[Kk][3] = {};   // acc[k][s] : 16n x 16ct tile, ct-subtile s of this wave

    const float* xb = x   + (size_t)b * Nn * CT;
    const float* sb = sat + (size_t)b * Nn * Nn;

    for (int m0 = 0; m0 < Nn; m0 += MB) {
        __syncthreads();   // previous iteration done reading LDS

        // ---- cooperative load: x chunk [MB][576] as float4 ----
        {
            const float4* src = (const float4*)(xb + (size_t)m0 * CT);
            for (int e = tid; e < MB * (CT / 4); e += BDIM) {
                int row = e / (CT / 4);
                int c4  = e % (CT / 4);
                float4 v = src[row * (CT / 4) + c4];
                *(float4*)(big + row * XPITCH + c4 * 4) = v;
            }
        }
        // ---- fused elementwise: tat[k][nn][mm] = cheb[k][m][n] * SA[b][m][n] ----
        for (int e = tid; e < MB * 16; e += BDIM) {
            int mm = e >> 4, nn = e & 15;
            float sv = sb[(size_t)(m0 + mm) * Nn + n0 + nn];
#pragma unroll
            for (int k = 0; k < Kk; ++k)
                tat[(k * 16 + nn) * TAT_PITCH + mm] =
                    cheb[((size_t)k * Nn + m0 + mm) * Nn + n0 + nn] * sv;
        }
        __syncthreads();

        // ---- stage 1 WMMA: rhs[n,ct] += t_at^T[n,m] * X[m,ct], 3 k in parallel ----
#pragma unroll
        for (int mb = 0; mb < MB / 4; ++mb) {
            const int ma = mb * 4 + half * 2;        // A/B frag K-base per lane half
            v2f afr[Kk];
#pragma unroll
            for (int k = 0; k < Kk; ++k)
                afr[k] = *(const v2f*)(tat + (k * 16 + nl) * TAT_PITCH + ma);
#pragma unroll
            for (int s = 0; s < 3; ++s) {
                const int ct0 = wave * 16 + s * 192;
                v2f bfr;
                bfr.x = big[(ma + 0) * XPITCH + ct0 + nl];
                bfr.y = big[(ma + 1) * XPITCH + ct0 + nl];
#pragma unroll
                for (int k = 0; k < Kk; ++k)
                    acc[k][s] = __builtin_amdgcn_wmma_f32_16x16x4_f32(
                        false, afr[k], false, bfr, (short)0, acc[k][s], false, false);
            }
        }
    }

    __syncthreads();   // done with X region; reuse as rhs
    // scatter accumulators: rhs[n][k*16+c][t]
#pragma unroll
    for (int k = 0; k < Kk; ++k)
#pragma unroll
        for (int s = 0; s < 3; ++s) {
            const int ct = wave * 16 + s * 192 + nl;
            const int c = ct / Tt, t = ct - c * Tt;
#pragma unroll
            for (int r = 0; r < 8; ++r) {
                int n = r + 8 * half;
                big[(n * 48 + k * 16 + c) * RHS_TP + t] = acc[k][s][r];
            }
        }
    __syncthreads();

    // ---- stage 2 WMMA: out[o,t] = relu( theta^T[o,kc] @ rhs[n][kc,t] ) ----
    for (int task = wave; task < 192; task += NWAVE) {   // uniform per wave: 16 tiles each
        const int n  = task & 15;
        const int j  = task >> 4;
        const int o0 = (j & 3) * 16;
        const int t0 = (j >> 2) * 16;
        v8f a2 = {};
#pragma unroll
        for (int ks = 0; ks < 12; ++ks) {
            v2f af = *(const v2f*)(thT + (o0 + nl) * TH_PITCH + ks * 4 + half * 2);
            v2f bf;
            bf.x = big[(n * 48 + ks * 4 + half * 2 + 0) * RHS_TP + t0 + nl];
            bf.y = big[(n * 48 + ks * 4 + half * 2 + 1) * RHS_TP + t0 + nl];
            a2 = __builtin_amdgcn_wmma_f32_16x16x4_f32(
                false, af, false, bf, (short)0, a2, false, false);
        }
        const int t = t0 + nl;
        if (t < Tt) {
            const size_t base = ((size_t)(b * Nn + n0 + n)) * Cout * Tt;
#pragma unroll
            for (int r = 0; r < 8; ++r) {
                int o = o0 + r + 8 * half;
                float v = a2[r];
                out[base + (size_t)o * Tt + t] = v > 0.f ? v : 0.f;
            }
        }
    }
}

extern "C" void kernel_launch(void* const* d_in, const int* in_sizes, int n_in,
                              void* d_out, int out_size, void* d_ws, size_t ws_size,
                              hipStream_t stream) {
    const float* x     = (const float*)d_in[0];  // [16,1024,16,36]
    const float* sat   = (const float*)d_in[1];  // [16,1024,1024]
    const float* cheb  = (const float*)d_in[2];  // [3,1024,1024]
    const float* theta = (const float*)d_in[3];  // [3,16,64]
    float* out = (float*)d_out;                  // [16,1024,64,36]

    dim3 grid(Bb * (Nn / 16));   // 1024 workgroups: (b, n-tile)
    dim3 block(BDIM);
    cheb_conv_sat_kernel<<<grid, block, LDS_BYTES, stream>>>(x, sat, cheb, theta, out);
}